// GCNEncoder_30700426232008
// MI455X (gfx1250) — compile-verified
//
#include <hip/hip_runtime.h>
#include <stdint.h>

#define GCN_IN 512
#define GCN_H  128

typedef __attribute__((ext_vector_type(2))) float v2f;
typedef __attribute__((ext_vector_type(8))) float v8f;

// ---------------- degree / normalization ----------------

__global__ void deg_init_kernel(int* __restrict__ deg, int n) {
    int i = blockIdx.x * blockDim.x + threadIdx.x;
    if (i < n) deg[i] = 1;  // self loop contributes 1 to every node's degree
}

__global__ void deg_count_kernel(const int* __restrict__ dst, int* __restrict__ deg, int e) {
    int i = blockIdx.x * blockDim.x + threadIdx.x;
    if (i < e) atomicAdd(&deg[dst[i]], 1);
}

__global__ void dinv_kernel(const int* __restrict__ deg, float* __restrict__ dinv, int n) {
    int i = blockIdx.x * blockDim.x + threadIdx.x;
    if (i < n) dinv[i] = rsqrtf((float)deg[i]);  // deg >= 1 always (self loop)
}

// ---------------- GEMM: xs = (feats @ W) * dinv[row]  via V_WMMA_F32_16X16X4_F32 ----------------
// Block = 256 threads = 8 waves. Block b covers rows [16b,16b+16); wave w covers cols [16w,16w+16).
// A frag (16x4 f32, 2 VGPRs/lane): lanes 0-15 hold {K0,K1} of row (lane&15); lanes 16-31 hold {K2,K3}.
// B frag mirrored: lanes 0-15 hold {B[K0][col],B[K1][col]}; lanes 16-31 hold {B[K2][col],B[K3][col]}.
// C/D (16x16 f32, 8 VGPRs): acc[r] is element (M = r + 8*(lane>>4), N = lane&15).

__global__ void __launch_bounds__(256)
gemm_wmma_kernel(const float* __restrict__ A, const float* __restrict__ B,
                 const float* __restrict__ dinv, float* __restrict__ xs, int n) {
    const int rowBase = blockIdx.x * 16;
    if (rowBase + 16 > n) return;                 // uniform per block: EXEC all-ones at WMMA
    const int lane    = threadIdx.x & 31;
    const int colBase = (threadIdx.x >> 5) * 16;  // wave id * 16
    const int mrow  = lane & 15;
    const int khalf = (lane >> 4) << 1;           // 0 or 2
    const int ncol  = lane & 15;

    const float* arow = A + (rowBase + mrow) * GCN_IN + khalf;      // 8B aligned
    const float* bcol = B + khalf * GCN_H + (colBase + ncol);

    v8f acc = (v8f)0.0f;
    #pragma unroll 4
    for (int k0 = 0; k0 < GCN_IN; k0 += 4) {
        v2f a = *(const v2f*)(arow + k0);         // {A[m][k0+kh], A[m][k0+kh+1]}
        v2f b;
        b.x = bcol[k0 * GCN_H];                   // B[k0+kh][col]
        b.y = bcol[(k0 + 1) * GCN_H];             // B[k0+kh+1][col]
        // (neg_a, A, neg_b, B, c_mod, C, reuse_a, reuse_b)
        acc = __builtin_amdgcn_wmma_f32_16x16x4_f32(false, a, false, b,
                                                    (short)0, acc, false, false);
    }

    const int rhalf = (lane >> 4) << 3;           // 0 or 8
    #pragma unroll
    for (int r = 0; r < 8; ++r) {
        const int row = rowBase + rhalf + r;
        xs[row * GCN_H + colBase + ncol] = acc[r] * dinv[row];
    }
}

// ---------------- self-loop init: out = xs ----------------

__global__ void self_copy_kernel(const float* __restrict__ xs, float* __restrict__ out, int count4) {
    int i = blockIdx.x * blockDim.x + threadIdx.x;
    if (i < count4) ((float4*)out)[i] = ((const float4*)xs)[i];
}

// ---------------- edge scatter: out[dst] += xs[src]  (one wave per edge) ----------------

__global__ void __launch_bounds__(256)
scatter_kernel(const float* __restrict__ xs, const int* __restrict__ src,
               const int* __restrict__ dst, float* __restrict__ out, int e) {
    const int wid  = (int)((blockIdx.x * (unsigned)blockDim.x + threadIdx.x) >> 5);
    const int lane = threadIdx.x & 31;
    if (wid >= e) return;
    const int s = src[wid];
    const int d = dst[wid];
    const float4 v = ((const float4*)(xs + s * GCN_H))[lane];   // 32 lanes x float4 = 128 cols
    float* ob = out + d * GCN_H + (lane << 2);
    atomicAdd(ob + 0, v.x);                                     // global_atomic_add_f32, L2-resident
    atomicAdd(ob + 1, v.y);
    atomicAdd(ob + 2, v.z);
    atomicAdd(ob + 3, v.w);
}

// ---------------- finalize: relu(out*dinv + b), JAX threefry dropout (key 42, p_keep=0.5) --------

__device__ __forceinline__ uint32_t rotl32(uint32_t x, int r) { return (x << r) | (x >> (32 - r)); }

__device__ __forceinline__ void threefry2x32(uint32_t k0, uint32_t k1, uint32_t& x0, uint32_t& x1) {
    const uint32_t k2 = k0 ^ k1 ^ 0x1BD11BDAu;
    x0 += k0; x1 += k1;
    x0 += x1; x1 = rotl32(x1, 13); x1 ^= x0;
    x0 += x1; x1 = rotl32(x1, 15); x1 ^= x0;
    x0 += x1; x1 = rotl32(x1, 26); x1 ^= x0;
    x0 += x1; x1 = rotl32(x1,  6); x1 ^= x0;
    x0 += k1; x1 += k2 + 1u;
    x0 += x1; x1 = rotl32(x1, 17); x1 ^= x0;
    x0 += x1; x1 = rotl32(x1, 29); x1 ^= x0;
    x0 += x1; x1 = rotl32(x1, 16); x1 ^= x0;
    x0 += x1; x1 = rotl32(x1, 24); x1 ^= x0;
    x0 += k2; x1 += k0 + 2u;
    x0 += x1; x1 = rotl32(x1, 13); x1 ^= x0;
    x0 += x1; x1 = rotl32(x1, 15); x1 ^= x0;
    x0 += x1; x1 = rotl32(x1, 26); x1 ^= x0;
    x0 += x1; x1 = rotl32(x1,  6); x1 ^= x0;
    x0 += k0; x1 += k1 + 3u;
    x0 += x1; x1 = rotl32(x1, 17); x1 ^= x0;
    x0 += x1; x1 = rotl32(x1, 29); x1 ^= x0;
    x0 += x1; x1 = rotl32(x1, 16); x1 ^= x0;
    x0 += x1; x1 = rotl32(x1, 24); x1 ^= x0;
    x0 += k1; x1 += k2 + 4u;
    x0 += x1; x1 = rotl32(x1, 13); x1 ^= x0;
    x0 += x1; x1 = rotl32(x1, 15); x1 ^= x0;
    x0 += x1; x1 = rotl32(x1, 26); x1 ^= x0;
    x0 += x1; x1 = rotl32(x1,  6); x1 ^= x0;
    x0 += k2; x1 += k0 + 5u;
}

__device__ __forceinline__ void finalize_elem(float* __restrict__ out, const float* __restrict__ dinv,
                                              const float* __restrict__ bias, int idx, uint32_t bits) {
    float v = out[idx] * dinv[idx >> 7] + bias[idx & (GCN_H - 1)];
    v = fmaxf(v, 0.0f);
    const float u = __uint_as_float((bits >> 9) | 0x3f800000u) - 1.0f;  // [0,1)
    out[idx] = (u < 0.5f) ? v * 2.0f : 0.0f;                            // keep -> h/0.5
}

__global__ void finalize_kernel(float* __restrict__ out, const float* __restrict__ dinv,
                                const float* __restrict__ bias, int half) {
    int i = blockIdx.x * blockDim.x + threadIdx.x;
    if (i >= half) return;
    uint32_t x0 = (uint32_t)i, x1 = (uint32_t)(i + half);   // JAX splits the iota in halves
    threefry2x32(0u, 42u, x0, x1);                          // jax.random.key(42) -> (0,42)
    finalize_elem(out, dinv, bias, i,        x0);
    finalize_elem(out, dinv, bias, i + half, x1);
}

// ---------------- launcher ----------------

extern "C" void kernel_launch(void* const* d_in, const int* in_sizes, int n_in,
                              void* d_out, int out_size, void* d_ws, size_t ws_size,
                              hipStream_t stream) {
    const float* feats = (const float*)d_in[0];   // [N, 512]
    const float* W     = (const float*)d_in[1];   // [512, 128]
    const float* bias  = (const float*)d_in[2];   // [128]
    const int*   edge  = (const int*)d_in[3];     // [2, E] (JAX x64 off -> int32)

    const int n = in_sizes[0] / GCN_IN;           // 50000
    const int e = in_sizes[3] / 2;                // 800000
    const int* src = edge;
    const int* dst = edge + e;

    char*  ws   = (char*)d_ws;
    int*   deg  = (int*)ws;                                   // 4*n bytes
    float* dinv = (float*)(ws + 4 * (size_t)n);               // 4*n bytes
    float* xs   = (float*)(ws + 8 * (size_t)n);               // n*128*4 bytes (16B aligned)
    float* out  = (float*)d_out;

    deg_init_kernel <<<(n + 255) / 256, 256, 0, stream>>>(deg, n);
    deg_count_kernel<<<(e + 255) / 256, 256, 0, stream>>>(dst, deg, e);
    dinv_kernel     <<<(n + 255) / 256, 256, 0, stream>>>(deg, dinv, n);

    gemm_wmma_kernel<<<n / 16, 256, 0, stream>>>(feats, W, dinv, xs, n);

    const int c4 = n * (GCN_H / 4);
    self_copy_kernel<<<(c4 + 255) / 256, 256, 0, stream>>>(xs, out, c4);

    const long long sthreads = (long long)e * 32;
    scatter_kernel<<<(int)((sthreads + 255) / 256), 256, 0, stream>>>(xs, src, dst, out, e);

    const int half = n * GCN_H / 2;
    finalize_kernel<<<(half + 255) / 256, 256, 0, stream>>>(out, dinv, bias, half);
}